// Encoder_11716670783825
// MI455X (gfx1250) — compile-verified
//
#include <hip/hip_runtime.h>
#include <hip/hip_bf16.h>
#include <math.h>

// ---------------------------------------------------------------------------
// Problem constants (from the reference)
// ---------------------------------------------------------------------------
#define NPTS   65536
#define KCLU   8192
#define DEG    32
#define OUTD   64
#define NB     64        // NPTS / BLOCK
#define BLOCK  1024

typedef __attribute__((ext_vector_type(2))) float v2f;
typedef __attribute__((ext_vector_type(8))) float v8f;

#define WMMA4(a, b, c) \
  __builtin_amdgcn_wmma_f32_16x16x4_f32(false, (a), false, (b), (short)0, (c), false, false)

#define NEG_INF (-3.402823466e38f)

// ---------------------------------------------------------------------------
// CDNA5 async global->LDS copy (tracked by ASYNCcnt, drained by s_wait_asynccnt)
// Low 32 bits of a generic pointer into LDS == wave-relative LDS byte offset
// (ISA 10.2 aperture mapping), which is what VDST of the async op expects.
// ---------------------------------------------------------------------------
__device__ __forceinline__ void async_lds_copy_b128(float* lds_dst, const float* gsrc) {
  unsigned ldsoff = (unsigned)(unsigned long long)lds_dst;
  asm volatile("global_load_async_to_lds_b128 %0, %1, off"
               :: "v"(ldsoff), "v"(gsrc)
               : "memory");
}
__device__ __forceinline__ void async_wait_all() {
  asm volatile("s_wait_asynccnt 0x0" ::: "memory");
}
// nfloats must be a multiple of 4; 16B-aligned src/dst
__device__ __forceinline__ void copy_lds_async(float* dst, const float* src, int nfloats) {
  for (int i4 = threadIdx.x; i4 < (nfloats >> 2); i4 += blockDim.x)
    async_lds_copy_b128(dst + i4 * 4, src + i4 * 4);
}
__device__ __forceinline__ void copy_lds(float* dst, const float* src, int n) {
  for (int i = threadIdx.x; i < n; i += blockDim.x) dst[i] = src[i];
}

// ---------------------------------------------------------------------------
// ca_fusion kernel: one workgroup (512 thr = 16 waves) per 1024-row block.
// 4 sweeps, recomputing S = Xb @ D0^T tile-wise with WMMA f32 16x16x4.
// Row/col softmax stats live entirely in LDS.
// ---------------------------------------------------------------------------
__device__ __forceinline__ float xfeat(const float* __restrict__ x,
                                       const float* __restrict__ cw,
                                       int node, int j) {
  // x @ W1 + b1 ; W1 is (2,64) row major
  return x[node * 2 + 0] * cw[j] + x[node * 2 + 1] * cw[64 + j] + cw[128 + j];
}
__device__ __forceinline__ float dfeat(const float* __restrict__ depth,
                                       const float* __restrict__ cw,
                                       int node, int j) {
  // depth[:,None] @ W2 + b2 ; W2 is (1,64)
  return depth[node] * cw[192 + j] + cw[256 + j];
}

// compute 128x128 score tile: S = A(128x64) * B(128x64)^T, both LDS (stride 66),
// result to sS (stride 130). Whole 16-wave block participates.
__device__ __forceinline__ void score_tile_512(const float* __restrict__ sA,
                                               const float* __restrict__ sB,
                                               float* __restrict__ sS) {
  const int lane = threadIdx.x & 31;
  const int wv   = threadIdx.x >> 5;   // 0..15
  const int hh   = lane >> 4;
  const int ln   = lane & 15;
  for (int st = wv; st < 64; st += 16) {
    const int mi = st >> 3, ni = st & 7;
    v8f acc = {};
#pragma unroll
    for (int k0 = 0; k0 < 64; k0 += 4) {
      const int kb = k0 + hh * 2;
      v2f a = { sA[(mi * 16 + ln) * 66 + kb], sA[(mi * 16 + ln) * 66 + kb + 1] };
      v2f b = { sB[(ni * 16 + ln) * 66 + kb], sB[(ni * 16 + ln) * 66 + kb + 1] };
      acc = WMMA4(a, b, acc);
    }
#pragma unroll
    for (int r = 0; r < 8; ++r)
      sS[(mi * 16 + r + hh * 8) * 130 + ni * 16 + ln] = acc[r];
  }
}

__global__ void __launch_bounds__(512)
ca_fusion_kernel(const float* __restrict__ x,      // N x 2
                 const float* __restrict__ depth,  // N
                 const float* __restrict__ W1,     // 2 x 64
                 const float* __restrict__ b1,     // 64
                 const float* __restrict__ W2,     // 1 x 64
                 const float* __restrict__ b2,     // 64
                 float* __restrict__ fx,           // N x 64 (fused_x)
                 float* __restrict__ fd)           // N x 64 (fused_d)
{
  extern __shared__ float smem[];
  float* sA   = smem;                 // 128*66  X b-tile
  float* sB   = sA  + 128 * 66;       // 128*66  D0 c-tile  [c][k]
  float* sC2  = sB  + 128 * 66;       // 128*66  aux tile (Xc or Db)
  float* sS   = sC2 + 128 * 66;       // 128*130 scores / probs
  float* mrow = sS  + 128 * 130;      // 1024
  float* lrow = mrow + 1024;
  float* mcol = lrow + 1024;
  float* lcol = mcol + 1024;
  float* cw   = lcol + 1024;          // 320: W1 rows, b1, W2, b2

  const int blk  = blockIdx.x;        // 0..63
  const int tid  = threadIdx.x;
  const int lane = tid & 31, wv = tid >> 5, hh = lane >> 4, ln = lane & 15;

  for (int i = tid; i < 64; i += 512) {
    cw[i]       = W1[i];
    cw[64 + i]  = W1[64 + i];
    cw[128 + i] = b1[i];
    cw[192 + i] = W2[i];
    cw[256 + i] = b2[i];
  }
  for (int i = tid; i < 1024; i += 512) {
    mrow[i] = NEG_INF; lrow[i] = 0.f;
    mcol[i] = NEG_INF; lcol[i] = 0.f;
  }
  __syncthreads();

  // ---------------- Sweep 1: row/col maxima ----------------
  for (int bt = 0; bt < 8; ++bt) {
    for (int i = tid; i < 128 * 64; i += 512) {
      int r = i >> 6, c = i & 63;
      sA[r * 66 + c] = xfeat(x, cw, blk * 1024 + bt * 128 + r, c);
    }
    __syncthreads();
    for (int ct = 0; ct < 8; ++ct) {
      for (int i = tid; i < 128 * 64; i += 512) {
        int r = i >> 6, c = i & 63;
        sB[r * 66 + c] = dfeat(depth, cw, ct * 128 + r, c);
      }
      __syncthreads();
      score_tile_512(sA, sB, sS);
      __syncthreads();
      for (int r = tid; r < 128; r += 512) {
        float m = NEG_INF;
        for (int c = 0; c < 128; ++c) m = fmaxf(m, sS[r * 130 + c]);
        mrow[bt * 128 + r] = fmaxf(mrow[bt * 128 + r], m);
      }
      for (int c = tid; c < 128; c += 512) {
        float m = NEG_INF;
        for (int b = 0; b < 128; ++b) m = fmaxf(m, sS[b * 130 + c]);
        mcol[ct * 128 + c] = fmaxf(mcol[ct * 128 + c], m);
      }
      __syncthreads();
    }
  }

  // ---------------- Sweep 2: row/col exp-sums ----------------
  for (int bt = 0; bt < 8; ++bt) {
    for (int i = tid; i < 128 * 64; i += 512) {
      int r = i >> 6, c = i & 63;
      sA[r * 66 + c] = xfeat(x, cw, blk * 1024 + bt * 128 + r, c);
    }
    __syncthreads();
    for (int ct = 0; ct < 8; ++ct) {
      for (int i = tid; i < 128 * 64; i += 512) {
        int r = i >> 6, c = i & 63;
        sB[r * 66 + c] = dfeat(depth, cw, ct * 128 + r, c);
      }
      __syncthreads();
      score_tile_512(sA, sB, sS);
      __syncthreads();
      for (int r = tid; r < 128; r += 512) {
        float m = mrow[bt * 128 + r], s = 0.f;
        for (int c = 0; c < 128; ++c) s += __expf(sS[r * 130 + c] - m);
        lrow[bt * 128 + r] += s;
      }
      for (int c = tid; c < 128; c += 512) {
        float m = mcol[ct * 128 + c], s = 0.f;
        for (int b = 0; b < 128; ++b) s += __expf(sS[b * 130 + c] - m);
        lcol[ct * 128 + c] += s;
      }
      __syncthreads();
    }
  }

  // ---------------- Sweep 3a: fused_d[b,:] = sum_c att_a[b,c] * X[c,:] ------
  for (int bt = 0; bt < 8; ++bt) {
    for (int i = tid; i < 128 * 64; i += 512) {
      int r = i >> 6, c = i & 63;
      sA[r * 66 + c] = xfeat(x, cw, blk * 1024 + bt * 128 + r, c);
    }
    __syncthreads();
    v8f fac0 = {}, fac1 = {};
    for (int ct = 0; ct < 8; ++ct) {
      for (int i = tid; i < 128 * 64; i += 512) {
        int r = i >> 6, c = i & 63;
        sB[r * 66 + c]  = dfeat(depth, cw, ct * 128 + r, c);
        sC2[r * 66 + c] = xfeat(x, cw, blk * 1024 + ct * 128 + r, c);
      }
      __syncthreads();
      score_tile_512(sA, sB, sS);
      __syncthreads();
      for (int i = tid; i < 128 * 128; i += 512) {
        int b = i >> 7, c = i & 127;
        sS[b * 130 + c] =
            __expf(sS[b * 130 + c] - mrow[bt * 128 + b]) / lrow[bt * 128 + b];
      }
      __syncthreads();
#pragma unroll
      for (int t = 0; t < 2; ++t) {
        const int st = wv * 2 + t, mi = st >> 2, ni = st & 3;
        v8f acc = (t == 0) ? fac0 : fac1;
        for (int k0 = 0; k0 < 128; k0 += 4) {
          const int kb = k0 + hh * 2;
          v2f a = { sS[(mi * 16 + ln) * 130 + kb], sS[(mi * 16 + ln) * 130 + kb + 1] };
          v2f b = { sC2[kb * 66 + ni * 16 + ln],   sC2[(kb + 1) * 66 + ni * 16 + ln] };
          acc = WMMA4(a, b, acc);
        }
        if (t == 0) fac0 = acc; else fac1 = acc;
      }
      __syncthreads();
    }
#pragma unroll
    for (int t = 0; t < 2; ++t) {
      const int st = wv * 2 + t, mi = st >> 2, ni = st & 3;
      v8f acc = (t == 0) ? fac0 : fac1;
#pragma unroll
      for (int r = 0; r < 8; ++r) {
        int node = blk * 1024 + bt * 128 + mi * 16 + r + hh * 8;
        fd[node * 64 + ni * 16 + ln] = acc[r];
      }
    }
  }

  // ---------------- Sweep 3b: fused_x[c,:] = sum_b att_b[b,c] * D[b,:] ------
  for (int ct = 0; ct < 8; ++ct) {
    for (int i = tid; i < 128 * 64; i += 512) {
      int r = i >> 6, c = i & 63;
      sB[r * 66 + c] = dfeat(depth, cw, ct * 128 + r, c);
    }
    __syncthreads();
    v8f fac0 = {}, fac1 = {};
    for (int bt = 0; bt < 8; ++bt) {
      for (int i = tid; i < 128 * 64; i += 512) {
        int r = i >> 6, c = i & 63;
        sA[r * 66 + c]  = xfeat(x, cw, blk * 1024 + bt * 128 + r, c);
        sC2[r * 66 + c] = dfeat(depth, cw, blk * 1024 + bt * 128 + r, c);
      }
      __syncthreads();
      score_tile_512(sA, sB, sS);
      __syncthreads();
      for (int i = tid; i < 128 * 128; i += 512) {
        int b = i >> 7, c = i & 127;
        sS[b * 130 + c] =
            __expf(sS[b * 130 + c] - mcol[ct * 128 + c]) / lcol[ct * 128 + c];
      }
      __syncthreads();
#pragma unroll
      for (int t = 0; t < 2; ++t) {
        const int st = wv * 2 + t, mi = st >> 2, ni = st & 3;
        v8f acc = (t == 0) ? fac0 : fac1;
        for (int k0 = 0; k0 < 128; k0 += 4) {
          const int kb = k0 + hh * 2;
          // A[m=c][k=b] = P[b][c]  (transposed read of sS)
          v2f a = { sS[kb * 130 + mi * 16 + ln], sS[(kb + 1) * 130 + mi * 16 + ln] };
          v2f b = { sC2[kb * 66 + ni * 16 + ln], sC2[(kb + 1) * 66 + ni * 16 + ln] };
          acc = WMMA4(a, b, acc);
        }
        if (t == 0) fac0 = acc; else fac1 = acc;
      }
      __syncthreads();
    }
#pragma unroll
    for (int t = 0; t < 2; ++t) {
      const int st = wv * 2 + t, mi = st >> 2, ni = st & 3;
      v8f acc = (t == 0) ? fac0 : fac1;
#pragma unroll
      for (int r = 0; r < 8; ++r) {
        int node = blk * 1024 + ct * 128 + mi * 16 + r + hh * 8;
        fx[node * 64 + ni * 16 + ln] = acc[r];
      }
    }
  }
}

// ---------------------------------------------------------------------------
// Generic WMMA GEMM:  C(n x 128) = A(n x din) @ W(din x 128) [+ bias]
// A is fetched through a mode functor so concat / relu / diff / gather inputs
// never get materialized.  W tile is staged via CDNA5 async global->LDS.
// ---------------------------------------------------------------------------
enum AMode {
  A_RELU_CONCAT64 = 0,   // relu( j<64 ? p0[r,j] : p1[r,j-64] )   din=128
  A_DD            = 1,   // j<64 ? p0[r,j] : p1[r,j-64]-p0[r,j-64] din=128
  A_CONCAT_DIFF128= 2,   // j<128 ? p0[r,j] : p1[r,j-128]-p0[r,j-128] din=256
  A_PLAIN         = 3
};

__device__ __forceinline__ float fetchA(int row, int j, int mode, int din,
                                        const float* __restrict__ p0,
                                        const float* __restrict__ p1,
                                        const long long* __restrict__ rowidx) {
  long long r = rowidx ? rowidx[row] : (long long)row;
  if (mode == A_RELU_CONCAT64) {
    float v = (j < 64) ? p0[r * 64 + j] : p1[r * 64 + (j - 64)];
    return fmaxf(v, 0.f);
  } else if (mode == A_DD) {
    return (j < 64) ? p0[r * 64 + j]
                    : (p1[r * 64 + (j - 64)] - p0[r * 64 + (j - 64)]);
  } else if (mode == A_CONCAT_DIFF128) {
    return (j < 128) ? p0[r * 128 + j]
                     : (p1[r * 128 + (j - 128)] - p0[r * 128 + (j - 128)]);
  }
  return p0[r * (long long)din + j];
}

__global__ void __launch_bounds__(256)
gemm128_kernel(const float* __restrict__ p0, const float* __restrict__ p1,
               const long long* __restrict__ rowidx,
               const float* __restrict__ Wg,   // din x 128
               const float* __restrict__ bias, // 128 or null
               float* __restrict__ C,          // n x 128
               int nrows, int din, int mode)
{
  extern __shared__ float smem[];
  float* sA = smem;            // 128 * 66
  float* sW = sA + 128 * 66;   // 64 * 128  (16B-aligned: 128*66*4 = 33792 B)

  const int tid = threadIdx.x;
  const int lane = tid & 31, wv = tid >> 5, hh = lane >> 4, ln = lane & 15;
  const int row0 = blockIdx.x * 128;

  v8f acc[8];
#pragma unroll
  for (int i = 0; i < 8; ++i) acc[i] = (v8f){};

  for (int kc = 0; kc < din; kc += 64) {
    // async-stage the contiguous 64x128 weight slab straight into LDS
    copy_lds_async(sW, Wg + (size_t)kc * 128, 64 * 128);
    // meanwhile synthesize the A tile through the mode functor
    for (int i = tid; i < 128 * 64; i += 256) {
      int r = i >> 6, c = i & 63;
      int row = row0 + r; if (row >= nrows) row = nrows - 1;
      sA[r * 66 + c] = fetchA(row, kc + c, mode, din, p0, p1, rowidx);
    }
    async_wait_all();
    __syncthreads();
#pragma unroll
    for (int k0 = 0; k0 < 64; k0 += 4) {
      const int kb = k0 + hh * 2;
      v2f b = { sW[kb * 128 + wv * 16 + ln], sW[(kb + 1) * 128 + wv * 16 + ln] };
#pragma unroll
      for (int mi = 0; mi < 8; ++mi) {
        v2f a = { sA[(mi * 16 + ln) * 66 + kb], sA[(mi * 16 + ln) * 66 + kb + 1] };
        acc[mi] = WMMA4(a, b, acc[mi]);
      }
    }
    __syncthreads();
  }

  const int n = wv * 16 + ln;
  const float bv = bias ? bias[n] : 0.f;
#pragma unroll
  for (int mi = 0; mi < 8; ++mi) {
#pragma unroll
    for (int r = 0; r < 8; ++r) {
      int m = row0 + mi * 16 + r + hh * 8;
      if (m < nrows) C[(long long)m * 128 + n] = acc[mi][r] + bv;
    }
  }
}

// ---------------------------------------------------------------------------
// Small head:  out[i,j] = [relu]( dot(relu?(H[i,:128]), W[:,j]) + b[j] ) *mask
// ---------------------------------------------------------------------------
__global__ void head_kernel(const float* __restrict__ H,
                            const float* __restrict__ W,   // 128 x dout
                            const float* __restrict__ b,   // dout
                            float* __restrict__ out,
                            int nrows, int dout, int relu_in, int relu_out,
                            const unsigned char* __restrict__ mask)
{
  long long i = (long long)blockIdx.x * blockDim.x + threadIdx.x;
  if (i >= (long long)nrows * dout) return;
  int row = (int)(i / dout), col = (int)(i % dout);
  float acc = b[col];
  for (int k = 0; k < 128; ++k) {
    float h = H[(long long)row * 128 + k];
    if (relu_in) h = fmaxf(h, 0.f);
    acc += h * W[k * dout + col];
  }
  if (relu_out) acc = fmaxf(acc, 0.f);
  if (mask) acc *= mask[row] ? 1.f : 0.f;
  out[i] = acc;
}

// ---------------------------------------------------------------------------
// Edge / cluster kernel: 32 contiguous edges per cluster, both pt_conv calls
// fused (64 rows). pos_nn / attn_nn weights resident in LDS (async-staged);
// MLP layers via WMMA; per-channel segment softmax over the 32 edges in LDS.
// ---------------------------------------------------------------------------
// M=64, N=128 GEMM from LDS A (stride lda) and LDS W (din x 128), 8 waves.
__device__ __forceinline__ void wmma_gemm64(const float* __restrict__ A, int lda,
                                            const float* __restrict__ W,
                                            const float* __restrict__ bias,
                                            float* __restrict__ Cst, int ldc,
                                            int Kdim, bool relu_out)
{
  const int lane = threadIdx.x & 31, wv = threadIdx.x >> 5;
  const int hh = lane >> 4, ln = lane & 15;
  for (int st = wv; st < 32; st += 8) {
    const int mi = st >> 3, ni = st & 7;
    v8f acc = {};
    for (int k0 = 0; k0 < Kdim; k0 += 4) {
      const int kb = k0 + hh * 2;
      v2f a = { A[(mi * 16 + ln) * lda + kb], A[(mi * 16 + ln) * lda + kb + 1] };
      v2f b = { W[kb * 128 + ni * 16 + ln],   W[(kb + 1) * 128 + ni * 16 + ln] };
      acc = WMMA4(a, b, acc);
    }
    const int n = ni * 16 + ln;
    const float bv = bias[n];
#pragma unroll
    for (int r = 0; r < 8; ++r) {
      float v = acc[r] + bv;
      if (relu_out) v = fmaxf(v, 0.f);
      Cst[(mi * 16 + r + hh * 8) * ldc + n] = v;
    }
  }
}

__global__ void __launch_bounds__(256)
edge_cluster_kernel(const long long* __restrict__ edge_src,   // E
                    const long long* __restrict__ centers,    // K
                    const float* __restrict__ pos_cur,        // N x 3
                    const float* __restrict__ pos_tar,        // N x 3
                    const float* __restrict__ xv_cur, const float* __restrict__ xv_tar,
                    const float* __restrict__ as_cur, const float* __restrict__ as_tar,
                    const float* __restrict__ a_dst,          // K x 128
                    const float* __restrict__ posW0, const float* __restrict__ posb0,
                    const float* __restrict__ posW1, const float* __restrict__ posb1,
                    const float* __restrict__ atW0,  const float* __restrict__ atb0,
                    const float* __restrict__ atW1,  const float* __restrict__ atb1,
                    float* __restrict__ clu_cur, float* __restrict__ clu_tar)
{
  extern __shared__ float smem[];
  float* wP0 = smem;               // 6*64          (off 0, 16B aligned)
  float* bP0 = wP0 + 384;          // 64
  float* wP1 = bP0 + 64;           // 64*128        (off 448*4 = 1792 B, aligned)
  float* bP1 = wP1 + 8192;         // 128
  float* wA0 = bP1 + 128;          // 128*128       (off 8768*4 = 35072 B, aligned)
  float* bA0 = wA0 + 16384;        // 128
  float* wA1 = bA0 + 128;          // 128*128       (off 25280*4 = 101120 B, aligned)
  float* bA1 = wA1 + 16384;        // 128
  float* sP6   = bA1 + 128;        // 64*8
  float* sH    = sP6 + 64 * 8;     // 64*66   relu(pos hidden)
  float* sDel  = sH + 64 * 66;     // 64*130  delta
  float* sAin  = sDel + 64 * 130;  // 64*130  alpha_in -> alpha
  float* sT    = sAin + 64 * 130;  // 64*130  relu(attn hidden)
  float* sAdC  = sT + 64 * 130;    // 128
  float* sStat = sAdC + 128;       // 512: [0:256) max, [256:512) sum
  int*   sSrc  = (int*)(sStat + 512); // 32

  const int tid = threadIdx.x;

  // async-stage the big weight matrices, plain-copy the small vectors
  copy_lds_async(wP0, posW0, 6 * 64);
  copy_lds_async(wP1, posW1, 64 * 128);
  copy_lds_async(wA0, atW0, 128 * 128);
  copy_lds_async(wA1, atW1, 128 * 128);
  copy_lds(bP0, posb0, 64);
  copy_lds(bP1, posb1, 128);
  copy_lds(bA0, atb0, 128);
  copy_lds(bA1, atb1, 128);
  async_wait_all();
  __syncthreads();

  for (int c = blockIdx.x; c < KCLU; c += gridDim.x) {
    const long long ctr = centers[c];
    if (tid < DEG) sSrc[tid] = (int)edge_src[(long long)c * DEG + tid];
    for (int i = tid; i < 128; i += 256) sAdC[i] = a_dst[(long long)c * 128 + i];
    __syncthreads();

    // pos inputs: row = call*32 + e, 6 dims (pos_dst - pos_src)
    for (int i = tid; i < 64 * 6; i += 256) {
      int row = i / 6, j = i % 6;
      int call = row >> 5, e = row & 31, jj = j % 3;
      int s = sSrc[e];
      const float* psrc = (j < 3) ? (call == 0 ? pos_cur : pos_tar) : pos_tar;
      sP6[row * 8 + j] = pos_tar[ctr * 3 + jj] - psrc[(long long)s * 3 + jj];
    }
    __syncthreads();

    // pos_nn layer0 (6 -> 64), store relu
    for (int i = tid; i < 64 * 64; i += 256) {
      int row = i >> 6, k = i & 63;
      float a = bP0[k];
#pragma unroll
      for (int j = 0; j < 6; ++j) a += sP6[row * 8 + j] * wP0[j * 64 + k];
      sH[row * 66 + k] = fmaxf(a, 0.f);
    }
    __syncthreads();

    // delta = relu(h) @ posW1 + b   (64x64x128, WMMA)
    wmma_gemm64(sH, 66, wP1, bP1, sDel, 130, 64, false);
    __syncthreads();

    // alpha_in = a_dst[c] - a_src[src] + delta
    for (int i = tid; i < 64 * 128; i += 256) {
      int row = i >> 7, ch = i & 127;
      int call = row >> 5, e = row & 31;
      int s = sSrc[e];
      const float* as = call ? as_tar : as_cur;
      sAin[row * 130 + ch] = sAdC[ch] - as[(long long)s * 128 + ch] + sDel[row * 130 + ch];
    }
    __syncthreads();

    // attn layer0 (store relu), then layer1 -> alpha (overwrite sAin)
    wmma_gemm64(sAin, 130, wA0, bA0, sT, 130, 128, true);
    __syncthreads();
    wmma_gemm64(sT, 130, wA1, bA1, sAin, 130, 128, false);
    __syncthreads();

    // per (call, channel) softmax over 32 edges
    for (int i = tid; i < 2 * 128; i += 256) {
      int call = i >> 7, ch = i & 127;
      float m = NEG_INF;
      for (int e = 0; e < DEG; ++e)
        m = fmaxf(m, sAin[(call * 32 + e) * 130 + ch]);
      if (!isfinite(m)) m = 0.f;
      float s = 0.f;
      for (int e = 0; e < DEG; ++e)
        s += __expf(sAin[(call * 32 + e) * 130 + ch] - m);
      sStat[i] = m;
      sStat[256 + i] = s;
    }
    __syncthreads();

    // message aggregation + final relu
    for (int i = tid; i < 2 * 128; i += 256) {
      int call = i >> 7, ch = i & 127;
      float m = sStat[i], s = sStat[256 + i];
      const float* xvp = call ? xv_tar : xv_cur;
      float acc = 0.f;
      for (int e = 0; e < DEG; ++e) {
        int row = call * 32 + e;
        float w = __expf(sAin[row * 130 + ch] - m) / (s + 1e-16f);
        acc += w * (xvp[(long long)sSrc[e] * 128 + ch] + sDel[row * 130 + ch]);
      }
      float o = fmaxf(acc, 0.f);
      if (call == 0) clu_cur[(long long)c * 128 + ch] = o;
      else           clu_tar[(long long)c * 128 + ch] = o;
    }
    __syncthreads();
  }
}

// ---------------------------------------------------------------------------
// Host side
// ---------------------------------------------------------------------------
extern "C" void kernel_launch(void* const* d_in, const int* in_sizes, int n_in,
                              void* d_out, int out_size, void* d_ws, size_t ws_size,
                              hipStream_t stream) {
  // Inputs in setup_inputs() dict order; params pytree in insertion order.
  const float*     x_cur     = (const float*)d_in[0];
  const float*     x_tar     = (const float*)d_in[1];
  const float*     pos_cur   = (const float*)d_in[2];
  const float*     pos_tar   = (const float*)d_in[3];
  const unsigned char* mask  = (const unsigned char*)d_in[4];
  const long long* edge_idx  = (const long long*)d_in[5];  // [2, E] -> src first
  const long long* centers   = (const long long*)d_in[6];
  const float*     depth_cur = (const float*)d_in[7];
  const float*     depth_tar = (const float*)d_in[8];
  const float* p1W  = (const float*)d_in[9];   // pathway1 (2,64)
  const float* p1b  = (const float*)d_in[10];
  const float* p2W  = (const float*)d_in[11];  // pathway2 (1,64)
  const float* p2b  = (const float*)d_in[12];
  const float* de0W = (const float*)d_in[13];  // depth_emb L0 (128,128)
  const float* de0b = (const float*)d_in[14];
  const float* de1W = (const float*)d_in[15];  // (128,6)
  const float* de1b = (const float*)d_in[16];
  const float* dn0W = (const float*)d_in[17];  // depth_emb_norm L0 (128,128)
  const float* dn0b = (const float*)d_in[18];
  const float* dn1W = (const float*)d_in[19];  // (128,1)
  const float* dn1b = (const float*)d_in[20];
  const float* linW  = (const float*)d_in[21]; // (128,128)
  const float* linSW = (const float*)d_in[22];
  const float* linDW = (const float*)d_in[23];
  const float* pn0W = (const float*)d_in[24];  // pos_nn (6,64)
  const float* pn0b = (const float*)d_in[25];
  const float* pn1W = (const float*)d_in[26];  // (64,128)
  const float* pn1b = (const float*)d_in[27];
  const float* at0W = (const float*)d_in[28];  // attn_nn (128,128)
  const float* at0b = (const float*)d_in[29];
  const float* at1W = (const float*)d_in[30];
  const float* at1b = (const float*)d_in[31];
  const float* oe0W = (const float*)d_in[32];  // out_enc (256,128)
  const float* oe0b = (const float*)d_in[33];
  const float* oe1W = (const float*)d_in[34];  // (128,64)
  const float* oe1b = (const float*)d_in[35];

  float* out = (float*)d_out;
  float* out_clu  = out;                              // K*64
  float* out_emb  = out + (size_t)KCLU * 64;          // N*6
  float* out_norm = out_emb + (size_t)NPTS * 6;       // N

  // ---- workspace layout (floats) ----
  float* ws = (float*)d_ws;
  size_t o = 0;
  float* fx_cur = ws + o; o += (size_t)NPTS * 64;
  float* fd_cur = ws + o; o += (size_t)NPTS * 64;
  float* fx_tar = ws + o; o += (size_t)NPTS * 64;
  float* fd_tar = ws + o; o += (size_t)NPTS * 64;
  float* xv_cur = ws + o; o += (size_t)NPTS * 128;
  float* as_cur = ws + o; o += (size_t)NPTS * 128;
  float* xv_tar = ws + o; o += (size_t)NPTS * 128;
  float* as_tar = ws + o; o += (size_t)NPTS * 128;
  float* a_dstB = ws + o; o += (size_t)KCLU * 128;
  float* cluC   = ws + o; o += (size_t)KCLU * 128;
  float* cluT   = ws + o; o += (size_t)KCLU * 128;
  float* H2     = ws + o; o += (size_t)KCLU * 128;
  float* H      = xv_cur;  // depth hidden scratch, reused before xv GEMMs

  const size_t caSmem   = (3 * 128 * 66 + 128 * 130 + 4 * 1024 + 320) * sizeof(float);
  const size_t gemmSmem = (128 * 66 + 64 * 128) * sizeof(float);
  const size_t edgeSmem = (384 + 64 + 8192 + 128 + 16384 + 128 + 16384 + 128 +
                           64 * 8 + 64 * 66 + 3 * 64 * 130 + 128 + 512 + 32) * sizeof(float);

  (void)hipFuncSetAttribute((const void*)ca_fusion_kernel,
                            hipFuncAttributeMaxDynamicSharedMemorySize, (int)caSmem);
  (void)hipFuncSetAttribute((const void*)edge_cluster_kernel,
                            hipFuncAttributeMaxDynamicSharedMemorySize, (int)edgeSmem);
  (void)hipFuncSetAttribute((const void*)gemm128_kernel,
                            hipFuncAttributeMaxDynamicSharedMemorySize, (int)gemmSmem);

  // 1) ca_fusion for cur and tar
  ca_fusion_kernel<<<NB, 512, caSmem, stream>>>(x_cur, depth_cur, p1W, p1b, p2W, p2b,
                                                fx_cur, fd_cur);
  ca_fusion_kernel<<<NB, 512, caSmem, stream>>>(x_tar, depth_tar, p1W, p1b, p2W, p2b,
                                                fx_tar, fd_tar);

  // 2) depth embedding heads (dd = concat(fd_cur, fd_tar - fd_cur))
  gemm128_kernel<<<NPTS / 128, 256, gemmSmem, stream>>>(fd_cur, fd_tar, nullptr,
                                                        de0W, de0b, H, NPTS, 128, A_DD);
  head_kernel<<<(NPTS * 6 + 255) / 256, 256, 0, stream>>>(H, de1W, de1b, out_emb,
                                                          NPTS, 6, 1, 0, nullptr);
  gemm128_kernel<<<NPTS / 128, 256, gemmSmem, stream>>>(fd_cur, fd_tar, nullptr,
                                                        dn0W, dn0b, H, NPTS, 128, A_DD);
  head_kernel<<<(NPTS + 255) / 256, 256, 0, stream>>>(H, dn1W, dn1b, out_norm,
                                                      NPTS, 1, 1, 0, nullptr);

  // 3) node-level projections: xv / a_src for both datasets, a_dst for centers
  gemm128_kernel<<<NPTS / 128, 256, gemmSmem, stream>>>(fx_cur, fd_cur, nullptr,
                                                        linW, nullptr, xv_cur, NPTS, 128,
                                                        A_RELU_CONCAT64);
  gemm128_kernel<<<NPTS / 128, 256, gemmSmem, stream>>>(fx_cur, fd_cur, nullptr,
                                                        linSW, nullptr, as_cur, NPTS, 128,
                                                        A_RELU_CONCAT64);
  gemm128_kernel<<<NPTS / 128, 256, gemmSmem, stream>>>(fx_tar, fd_tar, nullptr,
                                                        linW, nullptr, xv_tar, NPTS, 128,
                                                        A_RELU_CONCAT64);
  gemm128_kernel<<<NPTS / 128, 256, gemmSmem, stream>>>(fx_tar, fd_tar, nullptr,
                                                        linSW, nullptr, as_tar, NPTS, 128,
                                                        A_RELU_CONCAT64);
  gemm128_kernel<<<KCLU / 128, 256, gemmSmem, stream>>>(fx_tar, fd_tar, centers,
                                                        linDW, nullptr, a_dstB, KCLU, 128,
                                                        A_RELU_CONCAT64);

  // 4) fused per-cluster point-transformer conv (both calls)
  edge_cluster_kernel<<<512, 256, edgeSmem, stream>>>(
      edge_idx, centers, pos_cur, pos_tar, xv_cur, xv_tar, as_cur, as_tar, a_dstB,
      pn0W, pn0b, pn1W, pn1b, at0W, at0b, at1W, at1b, cluC, cluT);

  // 5) out_enc: concat(cluC, cluT - cluC) -> 128 hidden -> 64 out * mask
  gemm128_kernel<<<KCLU / 128, 256, gemmSmem, stream>>>(cluC, cluT, nullptr,
                                                        oe0W, oe0b, H2, KCLU, 256,
                                                        A_CONCAT_DIFF128);
  head_kernel<<<(KCLU * 64 + 255) / 256, 256, 0, stream>>>(H2, oe1W, oe1b, out_clu,
                                                           KCLU, 64, 1, 1, mask);
  (void)in_sizes; (void)n_in; (void)out_size; (void)ws_size;
}